// GNNLayer_27625229648304
// MI455X (gfx1250) — compile-verified
//
#include <hip/hip_runtime.h>

// -------- WMMA / TDM vector types (CDNA5 / gfx1250, wave32) --------
typedef __attribute__((ext_vector_type(2))) float v2f;
typedef __attribute__((ext_vector_type(8))) float v8f;
typedef __attribute__((ext_vector_type(4))) unsigned int u32x4;
typedef __attribute__((ext_vector_type(8))) int i32x8;
typedef __attribute__((ext_vector_type(4))) int i32x4;

#define D_IN   512
#define D_OUT  512
#define TILE_M 128   // block M tile (8 waves x 16 rows)
#define TILE_N 64    // block N tile (4 WMMA tiles per wave)
#define KC     64    // K chunk staged in LDS
#define KSTEP  (KC / 4)   // 16 WMMA k-steps per chunk
#define BS_STR 66    // LDS row stride in floats (64 + 2 pad, done by TDM pad)

#if defined(__has_builtin)
#if __has_builtin(__builtin_amdgcn_tensor_load_to_lds)
#define HAVE_TDM 1
#endif
#endif

// ---------------------------------------------------------------------------
// Kernel 1: zero the output accumulator (d_out is poisoned by the harness)
// ---------------------------------------------------------------------------
__global__ __launch_bounds__(256) void zero_f4(float4* __restrict__ p, long n4) {
  long i = (long)blockIdx.x * blockDim.x + threadIdx.x;
  if (i < n4) p[i] = make_float4(0.f, 0.f, 0.f, 0.f);
}

#ifdef HAVE_TDM
// ---------------------------------------------------------------------------
// TDM descriptor: DMA a 64(k) x 64(n) fp32 tile of W (row stride 512) into
// LDS, row-major, padding +2 dwords after every 64 dwords -> LDS stride 66.
// ---------------------------------------------------------------------------
__device__ __forceinline__ void tdm_load_b_chunk(const float* gsrc, void* ldsdst) {
  unsigned long long ga = (unsigned long long)(uintptr_t)gsrc;
  unsigned lds = (unsigned)(uintptr_t)ldsdst;

  u32x4 g0;
  g0[0] = 1u;                                   // count=1 valid descriptor
  g0[1] = lds;                                  // lds_addr (bytes)
  g0[2] = (unsigned)(ga & 0xFFFFFFFFu);         // global_addr[31:0]
  g0[3] = (unsigned)((ga >> 32) & 0x1FFFFFFu)   // global_addr[56:32]
        | (2u << 30);                           // type = 2 ("image")

  i32x8 g1;
  g1[0] = (2 << 16)                             // data_size = 4 bytes
        | (1 << 20)                             // pad_enable
        | (5 << 22)                             // pad_interval: 64 dwords
        | (1 << 25);                            // pad_amount:   2 dwords
  g1[1] = (int)(512u << 16);                    // tensor_dim0 = 512 (lo16)
  g1[2] = (int)(512u << 16);                    // dim0 hi=0 | tensor_dim1=512 lo16
  g1[3] = (int)(64u << 16);                     // dim1 hi=0 | tile_dim0 = 64
  g1[4] = 64;                                   // tile_dim1 = 64, tile_dim2 = 0
  g1[5] = 512;                                  // tensor_dim0_stride = 512 (lo32)
  g1[6] = 0;                                    // stride hi | tensor_dim1_stride
  g1[7] = 0;

  i32x4 z = {0, 0, 0, 0};
#if __clang_major__ >= 23
  i32x8 z8 = {0, 0, 0, 0, 0, 0, 0, 0};
  __builtin_amdgcn_tensor_load_to_lds(g0, g1, z, z, z8, 0);
#else
  __builtin_amdgcn_tensor_load_to_lds(g0, g1, z, z, 0);
#endif
}
#endif

// ---------------------------------------------------------------------------
// Kernel 2: support = features @ weight  via V_WMMA_F32_16X16X4_F32
//   grid  = (ceil(N/128), D_OUT/64), block = 256 threads = 8 waves
//   Weight K-chunks are DMA'd global->LDS by the Tensor Data Mover (wave 0),
//   double-buffered against the WMMA compute of the previous chunk.
//   A fragments for a whole chunk are staged in registers (16 x b64 clause).
// ---------------------------------------------------------------------------
__global__ __launch_bounds__(256) void gemm_wmma_f32(
    const float* __restrict__ A,   // [n_rows, 512]
    const float* __restrict__ W,   // [512, 512]
    float* __restrict__ S,         // [n_rows, 512]
    int n_rows) {
  __shared__ float Bs[2][KC * BS_STR];     // 2 x 16896 B, row-major Bs[k][n]

  const int bm   = blockIdx.x * TILE_M;
  const int bn   = blockIdx.y * TILE_N;
  const int wv   = threadIdx.x >> 5;       // wave id 0..7
  const int lane = threadIdx.x & 31;
  const int half = lane >> 4;              // 0: K pair {0,1}, 1: K pair {2,3}
  const int nl   = lane & 15;              // M index (A) / N index (B,C)

  int rowA = bm + wv * 16 + nl;            // clamped; stores bounds-guarded
  if (rowA >= n_rows) rowA = n_rows - 1;
  const float* Arow = A + (long)rowA * D_IN;

  v8f acc[4];
  #pragma unroll
  for (int nt = 0; nt < 4; ++nt)
    #pragma unroll
    for (int v = 0; v < 8; ++v) acc[nt][v] = 0.0f;

  const int NCHUNK = D_IN / KC;

#ifdef HAVE_TDM
  if (wv == 0) tdm_load_b_chunk(&W[bn], &Bs[0][0]);          // prologue DMA
#endif

  for (int c = 0; c < NCHUNK; ++c) {
    const int k0 = c * KC;

    // ---- stage the whole A chunk in registers: one 16 x b64 load clause ----
    v2f afrag[KSTEP];
    #pragma unroll
    for (int kk = 0; kk < KSTEP; ++kk)
      afrag[kk] = *(const v2f*)&Arow[k0 + kk * 4 + 2 * half];

    // prefetch next A chunk into near caches (speculative, wrapped in-bounds)
    __builtin_prefetch(Arow + ((k0 + KC) & (D_IN - 1)), 0, 3);

#ifdef HAVE_TDM
    if (wv == 0) __builtin_amdgcn_s_wait_tensorcnt(0);       // chunk c landed
    __syncthreads();
    if (wv == 0 && c + 1 < NCHUNK)                           // DMA chunk c+1
      tdm_load_b_chunk(&W[(long)(k0 + KC) * D_OUT + bn], &Bs[(c + 1) & 1][0]);
#else
    // ---- synchronous fallback: stage chunk into Bs[c&1], stride 66 ----
    {
      float* Bb = &Bs[c & 1][0];
      #pragma unroll
      for (int i = 0; i < 4; ++i) {
        int f    = threadIdx.x + i * 256;    // float4 index 0..1023
        int krow = f >> 4;
        int seg  = f & 15;
        float4 w4 = *(const float4*)&W[(long)(k0 + krow) * D_OUT + bn + seg * 4];
        float* d = &Bb[krow * BS_STR + seg * 4];
        d[0] = w4.x; d[1] = w4.y; d[2] = w4.z; d[3] = w4.w;
      }
    }
    __syncthreads();
#endif

    const float* Bbuf = &Bs[c & 1][0];

    // ---- 16 WMMA K-steps; A fragment reused across 4 N-tiles ----
    #pragma unroll
    for (int kk = 0; kk < KSTEP; ++kk) {
      // base of LDS row (kbase + 2*half), column nl
      const float* p = &Bbuf[(kk * 4 + 2 * half) * BS_STR + nl];
      #pragma unroll
      for (int nt = 0; nt < 4; ++nt) {
        // B 4x16 fp32 layout: b[v] = W[kbase + 2*half + v][n] (row-major LDS)
        v2f b;
        b[0] = p[nt * 16];
        b[1] = p[nt * 16 + BS_STR];
        acc[nt] = __builtin_amdgcn_wmma_f32_16x16x4_f32(
            /*neg_a=*/false, afrag[kk], /*neg_b=*/false, b,
            /*c_mod=*/(short)0, acc[nt], /*reuse_a=*/false, /*reuse_b=*/false);
      }
    }
    __syncthreads();   // all waves done reading buf before it is overwritten
  }

  // ---- epilogue: C/D layout lane<16 -> M=v, lane>=16 -> M=v+8; N=lane&15
  #pragma unroll
  for (int nt = 0; nt < 4; ++nt) {
    #pragma unroll
    for (int v = 0; v < 8; ++v) {
      int m    = v + 8 * half;
      int grow = bm + wv * 16 + m;
      if (grow < n_rows)
        S[(long)grow * D_OUT + bn + nt * 16 + nl] = acc[nt][v];
    }
  }
}

// ---------------------------------------------------------------------------
// Kernel 3: edge-parallel scatter: out[rows[e]] += vals[e] * support[cols[e]]
//   one wave32 per edge; 512 floats = 4 x (32 lanes x float4), coalesced
// ---------------------------------------------------------------------------
__global__ __launch_bounds__(256) void spmm_scatter(
    const float* __restrict__ S,
    const int* __restrict__ rows,
    const int* __restrict__ cols,
    const float* __restrict__ vals,
    float* __restrict__ out, int nE) {
  int e    = (int)(((long)blockIdx.x * blockDim.x + threadIdx.x) >> 5);
  int lane = threadIdx.x & 31;
  if (e >= nE) return;

  int   r = rows[e];
  int   c = cols[e];
  float v = vals[e];

  const float4* src = (const float4*)(S + (size_t)c * D_OUT);
  float*        dst = out + (size_t)r * D_OUT;

  #pragma unroll
  for (int j = 0; j < 4; ++j) {
    int f4 = lane + j * 32;          // 0..127
    float4 x = src[f4];
    int base = f4 * 4;
    atomicAdd(dst + base + 0, v * x.x);
    atomicAdd(dst + base + 1, v * x.y);
    atomicAdd(dst + base + 2, v * x.z);
    atomicAdd(dst + base + 3, v * x.w);
  }
}

// ---------------------------------------------------------------------------
// Kernel 4: in-place ReLU
// ---------------------------------------------------------------------------
__global__ __launch_bounds__(256) void relu_f4(float4* __restrict__ p, long n4) {
  long i = (long)blockIdx.x * blockDim.x + threadIdx.x;
  if (i < n4) {
    float4 x = p[i];
    x.x = fmaxf(x.x, 0.f); x.y = fmaxf(x.y, 0.f);
    x.z = fmaxf(x.z, 0.f); x.w = fmaxf(x.w, 0.f);
    p[i] = x;
  }
}

// ---------------------------------------------------------------------------
extern "C" void kernel_launch(void* const* d_in, const int* in_sizes, int n_in,
                              void* d_out, int out_size, void* d_ws, size_t ws_size,
                              hipStream_t stream) {
  const float* features = (const float*)d_in[0];   // [N, 512]
  const float* weight   = (const float*)d_in[1];   // [512, 512]
  const int*   rows     = (const int*)d_in[2];     // [E]
  const int*   cols     = (const int*)d_in[3];     // [E]
  const float* vals     = (const float*)d_in[4];   // [E]
  float*       out      = (float*)d_out;           // [N, 512]
  float*       support  = (float*)d_ws;            // [N, 512] scratch

  const int n_rows = in_sizes[0] / D_IN;
  const int nE     = in_sizes[2];
  const long n4    = (long)n_rows * D_OUT / 4;

  // 1) zero accumulator
  {
    long blocks = (n4 + 255) / 256;
    zero_f4<<<(unsigned)blocks, 256, 0, stream>>>((float4*)out, n4);
  }
  // 2) dense projection: support = features @ weight (WMMA f32, TDM staging)
  {
    dim3 grid((n_rows + TILE_M - 1) / TILE_M, D_OUT / TILE_N);
    gemm_wmma_f32<<<grid, 256, 0, stream>>>(features, weight, support, n_rows);
  }
  // 3) sparse aggregate: atomic scatter-add, one wave per edge
  {
    long threads = (long)nE * 32;
    long blocks  = (threads + 255) / 256;
    spmm_scatter<<<(unsigned)blocks, 256, 0, stream>>>(support, rows, cols, vals,
                                                       out, nE);
  }
  // 4) ReLU in place
  {
    long blocks = (n4 + 255) / 256;
    relu_f4<<<(unsigned)blocks, 256, 0, stream>>>((float4*)out, n4);
  }
}